// CausalSelfAttention_35115652612328
// MI455X (gfx1250) — compile-verified
//
#include <hip/hip_runtime.h>
#include <hip/hip_bf16.h>

// ---------------------------------------------------------------------------
// Decode-step causal self-attention with KV cache, tuned for MI455X (gfx1250).
// Memory-bound (~1.6 GB traffic, ~170 MFLOP): hipMemcpyAsync for the 512 MB
// cache copies, coalesced streaming for K/V, and fp32 WMMA
// (V_WMMA_F32_16X16X4_F32) for the four projections (exact fp32 precision).
// GEMV: K split across 8 waves/tile (LDS reduce) + dual accumulators + b128
// loads so the WMMA chain is short and the loads are 16B/lane contiguous.
// ---------------------------------------------------------------------------

#define D_MODEL   2048
#define NUM_HEADS 16
#define HEAD_DIM  128
#define MAX_SEQ   32768
#define NCHUNK    64
#define KSLICE    256   // K elements per wave in the GEMV (2048/8 waves)

typedef __attribute__((ext_vector_type(2))) float v2f;
typedef __attribute__((ext_vector_type(4))) float v4f;
typedef __attribute__((ext_vector_type(8))) float v8f;

// One wave accumulates a K-slice of 16 outputs y[n0..n0+15] of y = W*x
// (x @ W.T, row-major W) with V_WMMA_F32_16X16X4_F32.
// WMMA K slots form an unordered reduction, so we assign each lane a
// contiguous float4 of its row (b128 load) and feed .xy / .zw to two
// independent WMMAs: slots cover k..k+7 exactly once, A and B agree.
//   A 16x4: x broadcast into all 16 rows (M layout irrelevant)
//   B 4x16: lane&15 = output column, lane>>4 selects which K pair
//   D row 0 -> element 0 of the v8f on lanes 0..15.
__device__ __forceinline__ float gemv16_wave_wmma(const float* __restrict__ W,
                                                  const float* __restrict__ x,
                                                  int n0, int k0, int K) {
  const int lane = threadIdx.x & 31;
  const int half = lane >> 4;
  const int col  = lane & 15;
  const float* wp = W + (size_t)(n0 + col) * K + k0 + 4 * half;
  const float* xp = x + k0 + 4 * half;
  v8f acc0 = {}, acc1 = {};
  for (int k = 0; k < KSLICE; k += 8) {
    v4f a = *(const v4f*)(xp + k);      // global_load_b128
    v4f b = *(const v4f*)(wp + k);      // global_load_b128
    v2f a0 = {a.x, a.y}, a1 = {a.z, a.w};
    v2f b0 = {b.x, b.y}, b1 = {b.z, b.w};
    acc0 = __builtin_amdgcn_wmma_f32_16x16x4_f32(false, a0, false, b0,
                                                 (short)0, acc0, false, false);
    acc1 = __builtin_amdgcn_wmma_f32_16x16x4_f32(false, a1, false, b1,
                                                 (short)0, acc1, false, false);
  }
  return acc0[0] + acc1[0];   // D row 0; valid on lanes 0..15
}

// Block = 8 waves = one 16-output tile; wave w covers K slice [w*256, +256).
__device__ __forceinline__ void gemv16_block(const float* __restrict__ W,
                                             const float* __restrict__ x,
                                             float* __restrict__ y, int n0) {
  __shared__ float red[8][16];
  const int wave = threadIdx.x >> 5;
  const int lane = threadIdx.x & 31;
  float r = gemv16_wave_wmma(W, x, n0, wave * KSLICE, D_MODEL);
  if (lane < 16) red[wave][lane] = r;
  __syncthreads();
  if (threadIdx.x < 16) {
    float s = 0.f;
#pragma unroll
    for (int w = 0; w < 8; ++w) s += red[w][threadIdx.x];
    y[n0 + threadIdx.x] = s;
  }
}

// 384 blocks: block b -> matrix b/128 (Wq,Wk,Wv), tile (b%128)*16
__global__ __launch_bounds__(256) void qkv_gemv_kernel(
    const float* __restrict__ x,  const float* __restrict__ Wq,
    const float* __restrict__ Wk, const float* __restrict__ Wv,
    float* __restrict__ qkv) {
  const int mat  = (int)blockIdx.x >> 7;   // 0..2
  const int tile = (int)blockIdx.x & 127;
  const float* W = (mat == 0) ? Wq : (mat == 1) ? Wk : Wv;
  gemv16_block(W, x, qkv + mat * D_MODEL, tile * 16);
}

// 128 blocks: y = attn_out @ Wo.T
__global__ __launch_bounds__(256) void oproj_gemv_kernel(
    const float* __restrict__ attn, const float* __restrict__ Wo,
    float* __restrict__ y) {
  gemv16_block(Wo, attn, y, (int)blockIdx.x * 16);
}

// Write the freshly computed k,v row into the (already copied) caches.
__global__ __launch_bounds__(256) void scatter_kv_kernel(
    const float* __restrict__ qkv, const int* __restrict__ cip,
    float* __restrict__ out_k, float* __restrict__ out_v) {
  const int ci = *cip;
  const int t  = (int)(blockIdx.x * blockDim.x + threadIdx.x); // 0..2047
  out_k[(size_t)ci * D_MODEL + t] = qkv[D_MODEL + t];
  out_v[(size_t)ci * D_MODEL + t] = qkv[2 * D_MODEL + t];
}

// scores[h][j] = (q[h] . k[j][h]) / sqrt(128); one wave per key j.
__global__ __launch_bounds__(256) void scores_kernel(
    const float* __restrict__ qkv, const float* __restrict__ out_k,
    const int* __restrict__ cip, float* __restrict__ scores) {
  __shared__ float qs[D_MODEL];
  const int nk = *cip + 1;
  for (int i = threadIdx.x; i < D_MODEL; i += blockDim.x) qs[i] = qkv[i];
  __syncthreads();
  const int gw     = (int)((blockIdx.x * blockDim.x + threadIdx.x) >> 5);
  const int lane   = threadIdx.x & 31;
  const int nwaves = (int)((gridDim.x * blockDim.x) >> 5);
  const float scale = 0.08838834764831845f; // 1/sqrt(128)
  for (int j = gw; j < nk; j += nwaves) {
    const float* krow = out_k + (size_t)j * D_MODEL;
    for (int h = 0; h < NUM_HEADS; ++h) {
      v4f kv = *(const v4f*)(krow + h * HEAD_DIM + lane * 4);
      v4f qv = *(const v4f*)(qs   + h * HEAD_DIM + lane * 4);
      float p = kv.x * qv.x + kv.y * qv.y + kv.z * qv.z + kv.w * qv.w;
      for (int off = 16; off > 0; off >>= 1) p += __shfl_xor(p, off, 32);
      if (lane == 0) scores[(size_t)h * MAX_SEQ + j] = p * scale;
    }
  }
}

// Per-head softmax: max, exp-in-place, 1/sum. One block per head.
__global__ __launch_bounds__(256) void softmax_kernel(
    float* __restrict__ scores, const int* __restrict__ cip,
    float* __restrict__ inv_sum) {
  const int h  = blockIdx.x;
  const int nk = *cip + 1;
  float* s = scores + (size_t)h * MAX_SEQ;
  __shared__ float red[8];
  float m = -3.4e38f;
  for (int j = threadIdx.x; j < nk; j += 256) m = fmaxf(m, s[j]);
  for (int off = 16; off > 0; off >>= 1) m = fmaxf(m, __shfl_xor(m, off, 32));
  if ((threadIdx.x & 31) == 0) red[threadIdx.x >> 5] = m;
  __syncthreads();
  if (threadIdx.x == 0) {
    float mm = red[0];
    for (int i = 1; i < 8; ++i) mm = fmaxf(mm, red[i]);
    red[0] = mm;
  }
  __syncthreads();
  m = red[0];
  __syncthreads();
  float sum = 0.f;
  for (int j = threadIdx.x; j < nk; j += 256) {
    float e = expf(s[j] - m);
    s[j] = e;
    sum += e;
  }
  for (int off = 16; off > 0; off >>= 1) sum += __shfl_xor(sum, off, 32);
  if ((threadIdx.x & 31) == 0) red[threadIdx.x >> 5] = sum;
  __syncthreads();
  if (threadIdx.x == 0) {
    float t = 0.f;
    for (int i = 0; i < 8; ++i) t += red[i];
    inv_sum[h] = 1.0f / t;
  }
}

// Chunked weighted sum over V: partial[chunk][h][d] = sum_j p[h][j]*v[j][h][d]
__global__ __launch_bounds__(128) void attnv_kernel(
    const float* __restrict__ scores, const float* __restrict__ out_v,
    const int* __restrict__ cip, float* __restrict__ partial) {
  const int chunk = blockIdx.x;   // 0..NCHUNK-1
  const int h     = blockIdx.y;   // 0..15
  const int d     = threadIdx.x;  // 0..127
  const int nk    = *cip + 1;
  const int per   = MAX_SEQ / NCHUNK; // 512
  const int j0 = chunk * per;
  const int j1 = (j0 + per < nk) ? (j0 + per) : nk;
  float acc = 0.f;
  for (int j = j0; j < j1; ++j)
    acc += scores[(size_t)h * MAX_SEQ + j] *
           out_v[(size_t)j * D_MODEL + h * HEAD_DIM + d];
  partial[((size_t)chunk * NUM_HEADS + h) * HEAD_DIM + d] = acc;
}

__global__ __launch_bounds__(256) void attn_reduce_kernel(
    const float* __restrict__ partial, const float* __restrict__ inv_sum,
    float* __restrict__ attn_out) {
  const int t = (int)(blockIdx.x * blockDim.x + threadIdx.x); // 0..2047
  const int h = t >> 7;
  const int d = t & 127;
  float acc = 0.f;
  for (int c = 0; c < NCHUNK; ++c)
    acc += partial[((size_t)c * NUM_HEADS + h) * HEAD_DIM + d];
  attn_out[t] = acc * inv_sum[h];
}

extern "C" void kernel_launch(void* const* d_in, const int* in_sizes, int n_in,
                              void* d_out, int out_size, void* d_ws, size_t ws_size,
                              hipStream_t stream) {
  const float* x   = (const float*)d_in[0];
  const float* Wq  = (const float*)d_in[1];
  const float* Wk  = (const float*)d_in[2];
  const float* Wv  = (const float*)d_in[3];
  const float* Wo  = (const float*)d_in[4];
  const float* pk  = (const float*)d_in[5];
  const float* pv  = (const float*)d_in[6];
  const int*   cip = (const int*)d_in[7];

  // outputs, flat in return order: y[2048], new_k, new_v
  float* y     = (float*)d_out;
  float* out_k = y + D_MODEL;
  float* out_v = out_k + (size_t)MAX_SEQ * D_MODEL;

  // workspace layout (floats)
  float* ws       = (float*)d_ws;
  float* qkv      = ws;                              // 3*2048
  float* attn_out = ws + 3 * D_MODEL;                // 2048
  float* inv_sum  = ws + 4 * D_MODEL;                // 16
  float* scores   = ws + 16384;                      // 16*32768 (2 MB)
  float* partial  = scores + (size_t)NUM_HEADS * MAX_SEQ; // 64*2048 (512 KB)

  const size_t kvbytes = (size_t)MAX_SEQ * D_MODEL * sizeof(float);
  hipMemcpyAsync(out_k, pk, kvbytes, hipMemcpyDeviceToDevice, stream);
  hipMemcpyAsync(out_v, pv, kvbytes, hipMemcpyDeviceToDevice, stream);

  qkv_gemv_kernel<<<dim3(384), dim3(256), 0, stream>>>(x, Wq, Wk, Wv, qkv);
  scatter_kv_kernel<<<dim3(8), dim3(256), 0, stream>>>(qkv, cip, out_k, out_v);
  scores_kernel<<<dim3(512), dim3(256), 0, stream>>>(qkv, out_k, cip, scores);
  softmax_kernel<<<dim3(16), dim3(256), 0, stream>>>(scores, cip, inv_sum);
  attnv_kernel<<<dim3(NCHUNK, NUM_HEADS), dim3(128), 0, stream>>>(scores, out_v, cip, partial);
  attn_reduce_kernel<<<dim3(8), dim3(256), 0, stream>>>(partial, inv_sum, attn_out);
  oproj_gemv_kernel<<<dim3(128), dim3(256), 0, stream>>>(attn_out, Wo, y);
}